// NasAutoGraphCCell_36816459661706
// MI455X (gfx1250) — compile-verified
//
#include <hip/hip_runtime.h>

// Problem constants (from the reference)
#define N_NODES 50000
#define N_EDGES 800000
#define DIM     128
#define MTILES  (N_NODES / 16)       // 3125 (exact)
#define WAVES_PER_BLOCK 8

typedef float v2f __attribute__((ext_vector_type(2)));
typedef float v8f __attribute__((ext_vector_type(8)));

__device__ __forceinline__ float leaky01(float v) {
    return v > 0.0f ? v : 0.01f * v;
}

// ---------------------------------------------------------------------------
// Utility kernels
// ---------------------------------------------------------------------------
__global__ __launch_bounds__(256) void zero4_kernel(float4* __restrict__ p, int n4) {
    int i = blockIdx.x * blockDim.x + threadIdx.x;
    if (i < n4) p[i] = make_float4(0.f, 0.f, 0.f, 0.f);
}

__global__ __launch_bounds__(256) void copy4_kernel(const float4* __restrict__ s,
                                                    float4* __restrict__ d, int n4) {
    int i = blockIdx.x * blockDim.x + threadIdx.x;
    if (i < n4) d[i] = s[i];
}

// ---------------------------------------------------------------------------
// Plain GEMM: out[r][c] = sum_k A[r][k] * W[c][k] + bias[c]   (A @ W^T + b)
// One wave -> one 16-row tile, all 128 output columns (8 WMMA column tiles).
// A fragment (16x4 f32): lane L holds M = L%16, K = 2*(L/16) + {0,1}
// B fragment (4x16 f32): lane L holds N = L%16, K = 2*(L/16) + {0,1}
// D (16x16 f32, 8 VGPRs): VGPR j, lanes 0-15 -> M=j, lanes 16-31 -> M=j+8; N=L%16
// ---------------------------------------------------------------------------
__global__ __launch_bounds__(256) void gemm_bias_kernel(
    const float* __restrict__ A, const float* __restrict__ W,
    const float* __restrict__ bias, float* __restrict__ out)
{
    int lane = threadIdx.x & 31;
    int wave = threadIdx.x >> 5;
    int tile = blockIdx.x * WAVES_PER_BLOCK + wave;
    if (tile >= MTILES) return;

    int r16 = lane & 15;
    int kb  = (lane >> 4) << 1;              // 0 or 2
    const float* arow = A + (tile * 16 + r16) * DIM + kb;

    v8f acc[8] = {};
    for (int k = 0; k < DIM; k += 4) {
        v2f a = *(const v2f*)(arow + k);
#pragma unroll
        for (int n = 0; n < 8; ++n) {
            v2f b = *(const v2f*)(W + (n * 16 + r16) * DIM + k + kb);
            acc[n] = __builtin_amdgcn_wmma_f32_16x16x4_f32(
                false, a, false, b, (short)0, acc[n], false, false);
        }
    }

    int mb = (lane >> 4) << 3;               // 0 or 8
#pragma unroll
    for (int n = 0; n < 8; ++n) {
        float bn = bias[n * 16 + r16];
#pragma unroll
        for (int j = 0; j < 8; ++j) {
            out[(tile * 16 + mb + j) * DIM + n * 16 + r16] = acc[n][j] + bn;
        }
    }
}

// ---------------------------------------------------------------------------
// deg[src] += w ; cnt[dst] += 1   (one thread per edge, f32 atomics)
// ---------------------------------------------------------------------------
__global__ __launch_bounds__(256) void deg_cnt_kernel(
    const int* __restrict__ ei, const float* __restrict__ ew,
    float* __restrict__ deg, float* __restrict__ cnt)
{
    int e = blockIdx.x * blockDim.x + threadIdx.x;
    if (e >= N_EDGES) return;
    int src = ei[e];
    int dst = ei[N_EDGES + e];
    atomicAdd(&deg[src], ew[e]);
    atomicAdd(&cnt[dst], 1.0f);
}

__global__ __launch_bounds__(256) void dis_kernel(const float* __restrict__ deg,
                                                  float* __restrict__ dis)
{
    int i = blockIdx.x * blockDim.x + threadIdx.x;
    if (i >= N_NODES) return;
    float d = deg[i];
    dis[i] = d > 0.0f ? rsqrtf(d) : 0.0f;
}

// ---------------------------------------------------------------------------
// Fused edge scatter: one wave per edge, lane handles 4 features.
//   c1 = -dis[src]*w*dis[dst] ;  Tx1[dst]  += c1 * xh[src]
//                                msum[dst] += w  * xh[src]
// Single float4 gather of xh[src] feeds both scatters.
// ---------------------------------------------------------------------------
__global__ __launch_bounds__(256) void edge_scatter_kernel(
    const int* __restrict__ ei, const float* __restrict__ ew,
    const float* __restrict__ dis, const float* __restrict__ xh,
    float* __restrict__ tx1, float* __restrict__ msum)
{
    int gid  = blockIdx.x * blockDim.x + threadIdx.x;
    int e    = gid >> 5;
    int lane = threadIdx.x & 31;
    if (e >= N_EDGES) return;

    int   src = ei[e];
    int   dst = ei[N_EDGES + e];
    float w   = ew[e];
    float c1  = -dis[src] * w * dis[dst];

    const float4 v = *(const float4*)(xh + src * DIM + lane * 4);
    float* t = tx1  + dst * DIM + lane * 4;
    float* m = msum + dst * DIM + lane * 4;

    atomicAdd(t + 0, c1 * v.x);
    atomicAdd(t + 1, c1 * v.y);
    atomicAdd(t + 2, c1 * v.z);
    atomicAdd(t + 3, c1 * v.w);
    atomicAdd(m + 0, w * v.x);
    atomicAdd(m + 1, w * v.y);
    atomicAdd(m + 2, w * v.z);
    atomicAdd(m + 3, w * v.w);
}

// ---------------------------------------------------------------------------
// Fused second stage:
//   o1 = leaky(xh@Wc0^T + Tx1@Wc1^T + bc)
//   o2 = leaky((msum/max(cnt,1))@Wrel^T + xh@Wroot^T + brel)
//   s  = o1 + o2          (the mean division is folded into the A fragment)
// ---------------------------------------------------------------------------
__global__ __launch_bounds__(256) void fused_s_kernel(
    const float* __restrict__ xh,   const float* __restrict__ tx1,
    const float* __restrict__ msum, const float* __restrict__ cnt,
    const float* __restrict__ Wc0,  const float* __restrict__ Wc1,
    const float* __restrict__ Wrel, const float* __restrict__ Wroot,
    const float* __restrict__ bc,   const float* __restrict__ brel,
    float* __restrict__ s)
{
    int lane = threadIdx.x & 31;
    int wave = threadIdx.x >> 5;
    int tile = blockIdx.x * WAVES_PER_BLOCK + wave;
    if (tile >= MTILES) return;

    int r16 = lane & 15;
    int kb  = (lane >> 4) << 1;
    int row = tile * 16 + r16;

    float rc = 1.0f / fmaxf(cnt[row], 1.0f);   // mean scale for this lane's row

    const float* xrow = xh   + row * DIM + kb;
    const float* trow = tx1  + row * DIM + kb;
    const float* mrow = msum + row * DIM + kb;

    v8f acc1[8] = {};
    v8f acc2[8] = {};

    for (int k = 0; k < DIM; k += 4) {
        v2f ax = *(const v2f*)(xrow + k);
        v2f at = *(const v2f*)(trow + k);
        v2f am = *(const v2f*)(mrow + k);
        am.x *= rc; am.y *= rc;
#pragma unroll
        for (int n = 0; n < 8; ++n) {
            int wofs = (n * 16 + r16) * DIM + k + kb;
            v2f b0 = *(const v2f*)(Wc0   + wofs);
            v2f b1 = *(const v2f*)(Wc1   + wofs);
            v2f b2 = *(const v2f*)(Wrel  + wofs);
            v2f b3 = *(const v2f*)(Wroot + wofs);
            acc1[n] = __builtin_amdgcn_wmma_f32_16x16x4_f32(
                false, ax, false, b0, (short)0, acc1[n], false, false);
            acc1[n] = __builtin_amdgcn_wmma_f32_16x16x4_f32(
                false, at, false, b1, (short)0, acc1[n], false, false);
            acc2[n] = __builtin_amdgcn_wmma_f32_16x16x4_f32(
                false, am, false, b2, (short)0, acc2[n], false, false);
            acc2[n] = __builtin_amdgcn_wmma_f32_16x16x4_f32(
                false, ax, false, b3, (short)0, acc2[n], false, false);
        }
    }

    int mb = (lane >> 4) << 3;
#pragma unroll
    for (int n = 0; n < 8; ++n) {
        int   col = n * 16 + r16;
        float b1  = bc[col];
        float b2  = brel[col];
#pragma unroll
        for (int j = 0; j < 8; ++j) {
            float o1 = leaky01(acc1[n][j] + b1);
            float o2 = leaky01(acc2[n][j] + b2);
            s[(tile * 16 + mb + j) * DIM + col] = o1 + o2;
        }
    }
}

// ---------------------------------------------------------------------------
// Host-side launch
// ---------------------------------------------------------------------------
extern "C" void kernel_launch(void* const* d_in, const int* in_sizes, int n_in,
                              void* d_out, int out_size, void* d_ws, size_t ws_size,
                              hipStream_t stream) {
    (void)in_sizes; (void)n_in; (void)out_size; (void)ws_size;

    // Inputs (setup_inputs order)
    const float* x     = (const float*)d_in[1];   // d_in[0] = h (unused by reference)
    const int*   ei    = (const int*)  d_in[2];   // [2, E]
    const float* ew    = (const float*)d_in[3];
    const float* Wp    = (const float*)d_in[4];
    const float* bp    = (const float*)d_in[5];
    const float* Wc0   = (const float*)d_in[6];
    const float* Wc1   = (const float*)d_in[7];
    const float* bc    = (const float*)d_in[8];
    const float* Wrel  = (const float*)d_in[9];
    const float* brel  = (const float*)d_in[10];
    const float* Wroot = (const float*)d_in[11];
    const float* Wl    = (const float*)d_in[12];
    const float* bl    = (const float*)d_in[13];

    float* out = (float*)d_out;
    float* his = out;                    // [N,128]
    float* o3  = out + N_NODES * DIM;    // [N,128]  (also used in-place as s)

    // Workspace layout (floats)
    float* ws   = (float*)d_ws;
    float* xh   = ws;                          // N*128
    float* tx1  = xh  + N_NODES * DIM;         // N*128  (zeroed)
    float* msum = tx1 + N_NODES * DIM;         // N*128  (zeroed)
    float* deg  = msum + N_NODES * DIM;        // N      (zeroed)
    float* cnt  = deg + N_NODES;               // N      (zeroed)
    float* dis  = cnt + N_NODES;               // N

    // 1) zero the atomic-accumulated region: tx1, msum, deg, cnt (contiguous)
    {
        int n4 = (2 * N_NODES * DIM + 2 * N_NODES) / 4;
        zero4_kernel<<<(n4 + 255) / 256, 256, 0, stream>>>((float4*)tx1, n4);
    }

    // 2) his = x
    {
        int n4 = (N_NODES * DIM) / 4;
        copy4_kernel<<<(n4 + 255) / 256, 256, 0, stream>>>(
            (const float4*)x, (float4*)his, n4);
    }

    int gemm_blocks = (MTILES + WAVES_PER_BLOCK - 1) / WAVES_PER_BLOCK;

    // 3) xh = x @ Wp^T + bp
    gemm_bias_kernel<<<gemm_blocks, 256, 0, stream>>>(x, Wp, bp, xh);

    // 4) deg (by src), cnt (by dst)
    deg_cnt_kernel<<<(N_EDGES + 255) / 256, 256, 0, stream>>>(ei, ew, deg, cnt);

    // 5) dis = deg > 0 ? rsqrt(deg) : 0
    dis_kernel<<<(N_NODES + 255) / 256, 256, 0, stream>>>(deg, dis);

    // 6) fused edge scatter -> tx1, msum (one wave per edge)
    {
        long long threads = (long long)N_EDGES * 32;
        int blocks = (int)((threads + 255) / 256);
        edge_scatter_kernel<<<blocks, 256, 0, stream>>>(ei, ew, dis, xh, tx1, msum);
    }

    // 7) s = leaky(xh@Wc0^T + tx1@Wc1^T + bc) + leaky(mean@Wrel^T + xh@Wroot^T + brel)
    //    written into the o3 half of d_out
    fused_s_kernel<<<gemm_blocks, 256, 0, stream>>>(
        xh, tx1, msum, cnt, Wc0, Wc1, Wrel, Wroot, bc, brel, o3);

    // 8) o3 = s @ Wl^T + bl   (in place; each wave reads only its own row tile)
    gemm_bias_kernel<<<gemm_blocks, 256, 0, stream>>>(o3, Wl, bl, o3);
}